// RetinaNetLoss_38723425140853
// MI455X (gfx1250) — compile-verified
//
#include <hip/hip_runtime.h>
#include <math.h>

#define NCLS   80
#define NB     4
#define NA     100000
#define NM     32
#define IMGW   800.0f
#define IMGH   800.0f

typedef __attribute__((ext_vector_type(2))) float v2f;
typedef __attribute__((ext_vector_type(8))) float v8f;

// ---------------------------------------------------------------------------
// Wave32-wide sum via two V_WMMA_F32_16X16X4_F32 ops.
// Pass 1: A[m][k] holds partials (a.x per lane, a.y = 0), B = ones, C = 0.
//   D[m][n] = rowsum[m] = x[m] + x[m+16]  (A layout: lane L<16 -> (M=L,K=0),
//   lane L>=16 -> (M=L-16,K=2)). D layout gives each lane rowsum[0..7] or
//   rowsum[8..15] in its 8 accumulator VGPRs.
// In-lane add of the 8 accumulators -> half-sums; pass 2 repeats to combine
// the two halves, broadcasting the full 32-lane sum to every lane.
// EXEC must be all ones: call unconditionally from full blocks.
// ---------------------------------------------------------------------------
__device__ __forceinline__ float wave_sum_wmma(float x) {
  v2f a;    a[0] = x;    a[1] = 0.0f;
  v2f ones; ones[0] = 1.0f; ones[1] = 1.0f;
  v8f c = {};
  v8f d = __builtin_amdgcn_wmma_f32_16x16x4_f32(
      false, a, false, ones, (short)0, c, false, false);
  float t = d[0] + d[1] + d[2] + d[3] + d[4] + d[5] + d[6] + d[7];
  a[0] = t; a[1] = 0.0f;
  d = __builtin_amdgcn_wmma_f32_16x16x4_f32(
      false, a, false, ones, (short)0, c, false, false);
  return d[0];
}

// ---------------------------------------------------------------------------
// Kernel 0: zero the scalar accumulators in workspace.
// accum[0]=cls_sum, accum[1]=reg_sum, accum[2]=num_pos
// ---------------------------------------------------------------------------
__global__ void zero_accum_kernel(float* __restrict__ accum) {
  if (threadIdx.x < 4) accum[threadIdx.x] = 0.0f;
}

// ---------------------------------------------------------------------------
// Kernel 1: per-anchor target assignment + Huber reg loss + positive count.
// One thread per anchor; 32 GT boxes cached in LDS.
// Writes packed target (state code | best_class<<2) for the focal pass.
// ---------------------------------------------------------------------------
__global__ void targets_reg_kernel(const float* __restrict__ regression,
                                   const float* __restrict__ anchors,
                                   const float* __restrict__ annotations,
                                   int* __restrict__ targets,
                                   float* __restrict__ accum) {
  __shared__ float sann[NM * 5];
  __shared__ float sred[16];
  const int tid = threadIdx.x;
  const int b   = blockIdx.y;

  if (tid < NM * 5) sann[tid] = annotations[b * NM * 5 + tid];
  __syncthreads();

  const int a      = blockIdx.x * blockDim.x + tid;
  const bool active = (a < NA);
  const int ai     = active ? a : (NA - 1);

  const float4 anc = ((const float4*)anchors)[ai];
  const float aw = anc.z - anc.x;
  const float ah = anc.w - anc.y;
  const float area_a = aw * ah;

  float best = -1.0f;
  int   bj   = 0;
#pragma unroll
  for (int j = 0; j < NM; ++j) {
    const float bx1 = sann[j * 5 + 0], by1 = sann[j * 5 + 1];
    const float bx2 = sann[j * 5 + 2], by2 = sann[j * 5 + 3];
    const float ix1 = fmaxf(anc.x, bx1), iy1 = fmaxf(anc.y, by1);
    const float ix2 = fminf(anc.z, bx2), iy2 = fminf(anc.w, by2);
    const float inter = fmaxf(ix2 - ix1, 0.0f) * fmaxf(iy2 - iy1, 0.0f);
    const float area_b = (bx2 - bx1) * (by2 - by1);
    const float uni = fmaxf(area_a + area_b - inter, 1e-8f);
    const float iou = inter / uni;
    if (iou > best) { best = iou; bj = j; }  // strict > keeps first argmax
  }

  const bool pos = best >= 0.5f;
  const bool ign = (best > 0.4f) && !pos;
  const float cx = (anc.x + anc.z) * 0.5f;
  const float cy = (anc.y + anc.w) * 0.5f;
  const bool inside = (cx >= 0.0f) && (cx < IMGW) && (cy >= 0.0f) && (cy < IMGH);
  // state code: 0 = negative, 1 = positive, 2 = ignore (state == -1)
  const int code = inside ? (pos ? 1 : (ign ? 2 : 0)) : 2;
  const bool fpos = inside && pos;
  const int bcls = (int)sann[bj * 5 + 4];

  float reg_contrib = 0.0f;
  float pos_contrib = 0.0f;
  if (active) {
    targets[b * NA + a] = code | (bcls << 2);
    if (fpos) {
      pos_contrib = 1.0f;
      const float4 r = ((const float4*)regression)[b * NA + a];
      const float bx1 = sann[bj * 5 + 0], by1 = sann[bj * 5 + 1];
      const float bx2 = sann[bj * 5 + 2], by2 = sann[bj * 5 + 3];
      const float iw = 5.0f / aw;   // (1/aw) / REG_STD, REG_STD = 0.2
      const float ih = 5.0f / ah;
      const float t0 = (bx1 - anc.x) * iw;
      const float t1 = (by1 - anc.y) * ih;
      const float t2 = (bx2 - anc.z) * iw;
      const float t3 = (by2 - anc.w) * ih;
      const float d0 = fabsf(r.x - t0), d1 = fabsf(r.y - t1);
      const float d2 = fabsf(r.z - t2), d3 = fabsf(r.w - t3);
      const float thr = 1.0f / 9.0f;          // 1/sigma^2, sigma^2 = 9
      const float off = 0.5f / 9.0f;
      reg_contrib  = (d0 < thr) ? 4.5f * d0 * d0 : d0 - off;
      reg_contrib += (d1 < thr) ? 4.5f * d1 * d1 : d1 - off;
      reg_contrib += (d2 < thr) ? 4.5f * d2 * d2 : d2 - off;
      reg_contrib += (d3 < thr) ? 4.5f * d3 * d3 : d3 - off;
    }
  }

  // WMMA wave reductions (unconditional: EXEC all ones), then block reduce.
  const float wr = wave_sum_wmma(reg_contrib);
  const float wp = wave_sum_wmma(pos_contrib);
  const int lane = tid & 31;
  const int wid  = tid >> 5;
  if (lane == 0) { sred[wid] = wr; sred[8 + wid] = wp; }
  __syncthreads();
  if (tid == 0) {
    float rr = 0.0f, pp = 0.0f;
#pragma unroll
    for (int i = 0; i < 8; ++i) { rr += sred[i]; pp += sred[8 + i]; }
    atomicAdd(&accum[1], rr);
    atomicAdd(&accum[2], pp);
  }
}

// ---------------------------------------------------------------------------
// Kernel 2: focal loss over (B, A, 80) classification. Memory-bound stream:
// each thread handles one float4 (4 classes); 8M threads total, coalesced
// global_load_b128, hardware v_log_f32 via __logf.
// ---------------------------------------------------------------------------
__global__ void focal_kernel(const float* __restrict__ classification,
                             const int* __restrict__ targets,
                             float* __restrict__ accum) {
  __shared__ float sred[8];
  const int gi = blockIdx.x * 256 + threadIdx.x;      // < 8,000,000
  const int anchor = gi / 20;                         // b*NA + a  (80/4 = 20)
  const int q      = gi - anchor * 20;

  const float4 x = ((const float4*)classification)[gi];
  const int packed = targets[anchor];
  const int code   = packed & 3;
  const int bcls   = packed >> 2;

  float sum = 0.0f;
  if (code != 2) {                                    // valid (state != -1)
    const bool ispos = (code == 1);
    const int cbase = q * 4;
    const float xs[4] = {x.x, x.y, x.z, x.w};
#pragma unroll
    for (int k = 0; k < 4; ++k) {
      const float p = fminf(fmaxf(xs[k], 1.0e-4f), 1.0f - 1.0e-4f);
      const bool lab = ispos && (cbase + k == bcls);
      const float qv    = lab ? p : 1.0f - p;         // prob of the target
      const float alpha = lab ? 0.25f : 0.75f;
      const float om    = 1.0f - qv;
      sum += alpha * om * om * (-__logf(qv));         // alpha*(1-q)^2 * -log q
    }
  }

  const float w = wave_sum_wmma(sum);                 // unconditional
  const int lane = threadIdx.x & 31;
  const int wid  = threadIdx.x >> 5;
  if (lane == 0) sred[wid] = w;
  __syncthreads();
  if (threadIdx.x == 0) {
    float r = sred[0] + sred[1] + sred[2] + sred[3] +
              sred[4] + sred[5] + sred[6] + sred[7];
    atomicAdd(&accum[0], r);
  }
}

// ---------------------------------------------------------------------------
// Kernel 3: final scalar.
// ---------------------------------------------------------------------------
__global__ void finalize_kernel(const float* __restrict__ accum,
                                float* __restrict__ out) {
  if (threadIdx.x == 0) {
    const float np = fmaxf(accum[2], 1.0f);
    out[0] = (accum[0] + accum[1]) / np;
  }
}

// ---------------------------------------------------------------------------
extern "C" void kernel_launch(void* const* d_in, const int* in_sizes, int n_in,
                              void* d_out, int out_size, void* d_ws, size_t ws_size,
                              hipStream_t stream) {
  const float* classification = (const float*)d_in[0];  // (B, A, 80)
  const float* regression     = (const float*)d_in[1];  // (B, A, 4)
  const float* anchors        = (const float*)d_in[2];  // (A, 4)
  const float* annotations    = (const float*)d_in[3];  // (B, M, 5)

  float* accum   = (float*)d_ws;                        // 4 floats
  int*   targets = (int*)((char*)d_ws + 256);           // B*A ints (1.6 MB)

  zero_accum_kernel<<<1, 32, 0, stream>>>(accum);

  dim3 grid1((NA + 255) / 256, NB);
  targets_reg_kernel<<<grid1, 256, 0, stream>>>(regression, anchors,
                                                annotations, targets, accum);

  const int nfloat4 = NB * NA * (NCLS / 4);             // 8,000,000
  focal_kernel<<<nfloat4 / 256, 256, 0, stream>>>(classification, targets,
                                                  accum);

  finalize_kernel<<<1, 32, 0, stream>>>(accum, (float*)d_out);
}